// MultilevelDetectionGenerator_84250078478552
// MI455X (gfx1250) — compile-verified
//
#include <hip/hip_runtime.h>
#include <math.h>

#define B_      4
#define N_ANCH  76800
#define NCLS    80
#define KTOP    512
#define MAXDET  100
#define FK      (NCLS * KTOP)   // 40960
#define SCORE_TH 0.05f

typedef __attribute__((ext_vector_type(2))) float v2f;
typedef __attribute__((ext_vector_type(8))) float v8f;

typedef int v4i __attribute__((vector_size(16)));
typedef __attribute__((address_space(1))) v4i* gasync_t;   // global src for async-to-LDS
typedef __attribute__((address_space(3))) v4i* lasync_t;   // LDS dst for async-to-LDS

// ---------------------------------------------------------------------------
// Kernel 0: transpose scores (B,N,C) -> (B,C,N) for coalesced radix scans
// ---------------------------------------------------------------------------
__global__ __launch_bounds__(256) void transpose_kernel(const float* __restrict__ in,
                                                        float* __restrict__ out) {
    __shared__ float tile[32][33];
    int b  = blockIdx.z;
    int n0 = blockIdx.x * 32;
    int c0 = blockIdx.y * 32;
    int tx = threadIdx.x & 31;
    int ty = threadIdx.x >> 5;   // 8 rows of 32
    for (int r = ty; r < 32; r += 8) {
        int n = n0 + r, c = c0 + tx;
        tile[r][tx] = (n < N_ANCH && c < NCLS) ? in[((size_t)b * N_ANCH + n) * NCLS + c] : 0.f;
    }
    __syncthreads();
    for (int r = ty; r < 32; r += 8) {
        int c = c0 + r, n = n0 + tx;
        if (c < NCLS && n < N_ANCH)
            out[((size_t)b * NCLS + c) * N_ANCH + n] = tile[tx][r];
    }
}

// ---------------------------------------------------------------------------
// Kernel 1: per (b,c) exact top-512 (threshold + stable descending order)
// 4-round radix select on float-ordered uint keys, compaction, bitonic sort.
// ---------------------------------------------------------------------------
__global__ __launch_bounds__(256) void topk_kernel(const float* __restrict__ scores_base,
                                                   int strideN,
                                                   const float* __restrict__ boxes,
                                                   float* __restrict__ top_scores,
                                                   float* __restrict__ top_boxes) {
    int bc = blockIdx.x;
    int b  = bc / NCLS;
    int c  = bc % NCLS;
    const float* sp = (strideN == 1)
        ? scores_base + (size_t)(b * NCLS + c) * N_ANCH
        : scores_base + (size_t)b * N_ANCH * NCLS + c;
    size_t sstride = (strideN == 1) ? 1 : (size_t)NCLS;

    __shared__ unsigned hist[256];
    __shared__ unsigned sPrefix, sRemaining, sCount;
    __shared__ unsigned long long kv[KTOP];
    int tid = threadIdx.x;

    if (tid == 0) { sPrefix = 0u; sRemaining = KTOP; }
    __syncthreads();

    // radix select the exact 512th-largest key T
    for (int round = 0; round < 4; ++round) {
        int shift = 24 - 8 * round;
        unsigned prefmask = (round == 0) ? 0u : (0xFFFFFFFFu << (shift + 8));
        for (int i = tid; i < 256; i += 256) hist[i] = 0u;
        __syncthreads();
        unsigned pref = sPrefix;
        for (int n = tid; n < N_ANCH; n += 256) {
            float s = sp[(size_t)n * sstride];
            unsigned key = (s > SCORE_TH) ? __float_as_uint(s) : 0u;
            if ((key & prefmask) == pref) atomicAdd(&hist[(key >> shift) & 0xFF], 1u);
        }
        __syncthreads();
        if (tid == 0) {
            unsigned rem = sRemaining, cum = 0u;
            int bin = 0;
            for (int bq = 255; bq >= 0; --bq) {
                if (cum + hist[bq] >= rem) { bin = bq; break; }
                cum += hist[bq];
            }
            sRemaining = (rem > cum) ? rem - cum : 1u;
            sPrefix = pref | ((unsigned)bin << shift);
        }
        __syncthreads();
    }
    unsigned T = sPrefix;

    if (tid == 0) sCount = 0u;
    __syncthreads();
    // compact strictly-greater keys (guaranteed < KTOP of them)
    for (int n = tid; n < N_ANCH; n += 256) {
        float s = sp[(size_t)n * sstride];
        unsigned key = (s > SCORE_TH) ? __float_as_uint(s) : 0u;
        if (key > T) {
            unsigned pos = atomicAdd(&sCount, 1u);
            if (pos < KTOP)
                kv[pos] = ((unsigned long long)key << 32) | (unsigned)(~(unsigned)n);
        }
    }
    __syncthreads();
    // fill remaining slots with ==T keys
    for (int n = tid; n < N_ANCH; n += 256) {
        float s = sp[(size_t)n * sstride];
        unsigned key = (s > SCORE_TH) ? __float_as_uint(s) : 0u;
        if (key == T) {
            unsigned pos = atomicAdd(&sCount, 1u);
            if (pos < KTOP)
                kv[pos] = ((unsigned long long)key << 32) | (unsigned)(~(unsigned)n);
        }
    }
    __syncthreads();
    unsigned total = sCount;
    if (total > KTOP) total = KTOP;
    for (unsigned i = total + tid; i < KTOP; i += 256) kv[i] = 0ull;
    __syncthreads();

    // bitonic sort, descending (key desc, then index asc via ~n in low bits)
    for (unsigned k = 2; k <= KTOP; k <<= 1) {
        for (unsigned j = k >> 1; j > 0; j >>= 1) {
            for (unsigned i = tid; i < KTOP; i += 256) {
                unsigned ixj = i ^ j;
                if (ixj > i) {
                    unsigned long long a = kv[i], bb = kv[ixj];
                    bool descRegion = ((i & k) == 0);
                    bool doSwap = descRegion ? (a < bb) : (a > bb);
                    if (doSwap) { kv[i] = bb; kv[ixj] = a; }
                }
            }
            __syncthreads();
        }
    }

    size_t obase = (size_t)bc * KTOP;
    for (int i = tid; i < KTOP; i += 256) {
        unsigned long long v = kv[i];
        unsigned key = (unsigned)(v >> 32);
        unsigned n   = ~((unsigned)v);
        if (key == 0u || n >= N_ANCH) n = 0u;
        float sc = key ? __uint_as_float(key) : -INFINITY;
        top_scores[obase + i] = sc;
        size_t bsrc = ((size_t)b * N_ANCH + n) * 4;
        size_t bdst = (obase + i) * 4;
        top_boxes[bdst + 0] = boxes[bsrc + 0];
        top_boxes[bdst + 1] = boxes[bsrc + 1];
        top_boxes[bdst + 2] = boxes[bsrc + 2];
        top_boxes[bdst + 3] = boxes[bsrc + 3];
    }
}

// ---------------------------------------------------------------------------
// Kernel 2: per (b,c) greedy NMS over 512 boxes.
// Box tile staged global->LDS with GLOBAL_LOAD_ASYNC_TO_LDS_B128 (ASYNCcnt).
// Pairwise area-sum (union term) computed with V_WMMA_F32_16X16X4_F32:
//   A row m = [area_m, 1, 0, 0], B col n = [1, area_n, 0, 0]  =>  D = area_m + area_n
// Suppression test: iou>0.5  <=>  3*inter > area_m + area_n.
// ---------------------------------------------------------------------------
__global__ __launch_bounds__(256) void nms_kernel(const float* __restrict__ top_scores,
                                                  const float* __restrict__ top_boxes,
                                                  float* __restrict__ cls_scores) {
    int bc  = blockIdx.x;
    int tid = threadIdx.x;
    __shared__ float sb[KTOP * 4];
    __shared__ float sarea[KTOP];
    __shared__ unsigned short smask[KTOP * 32];  // 512 rows x 32 col-tiles of 16 bits
    __shared__ unsigned ssup[16];
    __shared__ unsigned char skeep[KTOP];

    size_t base = (size_t)bc * KTOP;

#if __has_builtin(__builtin_amdgcn_global_load_async_to_lds_b128)
    // CDNA5 async DMA: global -> LDS without a VGPR round-trip (ASYNCcnt).
    for (int i = tid; i < KTOP; i += 256) {
        __builtin_amdgcn_global_load_async_to_lds_b128(
            (gasync_t)(top_boxes + (base + i) * 4),
            (lasync_t)&sb[i * 4], 0, 0);
    }
#if __has_builtin(__builtin_amdgcn_s_wait_asynccnt)
    __builtin_amdgcn_s_wait_asynccnt(0);
#else
    asm volatile("s_wait_asynccnt 0" ::: "memory");
#endif
    __syncthreads();
    for (int i = tid; i < KTOP; i += 256) {
        float y1 = sb[i * 4 + 0], x1 = sb[i * 4 + 1];
        float y2 = sb[i * 4 + 2], x2 = sb[i * 4 + 3];
        sarea[i] = (y2 - y1) * (x2 - x1);
    }
#else
    for (int i = tid; i < KTOP; i += 256) {
        float y1 = top_boxes[(base + i) * 4 + 0];
        float x1 = top_boxes[(base + i) * 4 + 1];
        float y2 = top_boxes[(base + i) * 4 + 2];
        float x2 = top_boxes[(base + i) * 4 + 3];
        sb[i * 4 + 0] = y1; sb[i * 4 + 1] = x1;
        sb[i * 4 + 2] = y2; sb[i * 4 + 3] = x2;
        sarea[i] = (y2 - y1) * (x2 - x1);
    }
#endif
    if (tid < 16) ssup[tid] = 0u;
    __syncthreads();

    int lane = tid & 31;
    int wave = tid >> 5;
    int half = lane >> 4;    // 0: lanes 0-15, 1: lanes 16-31
    int l16  = lane & 15;

    // 32x32 = 1024 tiles of 16x16, round-robin over 8 waves
    for (int t = wave; t < 1024; t += 8) {
        int tr = t >> 5;
        int tc = t & 31;

        float areaRow = sarea[tr * 16 + l16];
        float areaCol = sarea[tc * 16 + l16];
        v2f a, bm;
        a.x  = half ? 0.f : areaRow;   // A: K=0 (lanes0-15) / K=2 zero (lanes16-31)
        a.y  = half ? 0.f : 1.0f;      // A: K=1 / K=3
        bm.x = half ? 0.f : 1.0f;      // B: K=0 row / K=2 zero
        bm.y = half ? 0.f : areaCol;   // B: K=1 row / K=3 zero
        v8f cz = {};
#if __has_builtin(__builtin_amdgcn_wmma_f32_16x16x4_f32)
        v8f d = __builtin_amdgcn_wmma_f32_16x16x4_f32(false, a, false, bm,
                                                      (short)0, cz, false, false);
#else
        v8f d;
        #pragma unroll
        for (int r = 0; r < 8; ++r)
            d[r] = sarea[tr * 16 + r + (half ? 8 : 0)] + areaCol;
#endif
        int nIdx = tc * 16 + l16;
        float cy1 = sb[nIdx * 4 + 0], cx1 = sb[nIdx * 4 + 1];
        float cy2 = sb[nIdx * 4 + 2], cx2 = sb[nIdx * 4 + 3];
        #pragma unroll
        for (int r = 0; r < 8; ++r) {
            int m = tr * 16 + r + (half ? 8 : 0);
            float ry1 = sb[m * 4 + 0], rx1 = sb[m * 4 + 1];
            float ry2 = sb[m * 4 + 2], rx2 = sb[m * 4 + 3];
            float iy1 = fmaxf(ry1, cy1), ix1 = fmaxf(rx1, cx1);
            float iy2 = fminf(ry2, cy2), ix2 = fminf(rx2, cx2);
            float inter = fmaxf(iy2 - iy1, 0.f) * fmaxf(ix2 - ix1, 0.f);
            bool cond = (3.0f * inter > d[r]);   // iou > 0.5
            unsigned msk = __builtin_amdgcn_ballot_w32(cond);
            if (lane == 0)  smask[(tr * 16 + r) * 32 + tc]     = (unsigned short)(msk & 0xFFFFu);
            if (lane == 16) smask[(tr * 16 + r + 8) * 32 + tc] = (unsigned short)(msk >> 16);
        }
    }
    __syncthreads();

    // greedy suppression sweep: wave 0, lanes 0-15 own one u32 word each
    if (wave == 0) {
        volatile unsigned* vsup = ssup;
        for (int i = 0; i < KTOP; ++i) {
            unsigned w = vsup[i >> 5];
            bool keep = ((w >> (i & 31)) & 1u) == 0u;
            if (keep && lane < 16) {
                unsigned row = (unsigned)smask[i * 32 + lane * 2]
                             | ((unsigned)smask[i * 32 + lane * 2 + 1] << 16);
                int iw = i >> 5, ib = i & 31;
                if (lane < iw)       row = 0u;
                else if (lane == iw) row &= (ib == 31) ? 0u : (~0u << (ib + 1));
                atomicOr(&ssup[lane], row);
            }
        }
    }
    __syncthreads();

    // finiteness + rank<=MAXDET
    if (tid == 0) {
        int cnt = 0;
        for (int i = 0; i < KTOP; ++i) {
            bool k = ((ssup[i >> 5] >> (i & 31)) & 1u) == 0u;
            float sc = top_scores[base + i];
            k = k && (sc != -INFINITY);
            if (k) { cnt++; if (cnt > MAXDET) k = false; }
            skeep[i] = k ? 1 : 0;
        }
    }
    __syncthreads();
    for (int i = tid; i < KTOP; i += 256) {
        float sc = top_scores[base + i];
        cls_scores[base + i] = skeep[i] ? sc : -INFINITY;
    }
}

// ---------------------------------------------------------------------------
// Kernel 3: per-batch stable top-100 over 40960 candidates + output assembly.
// ---------------------------------------------------------------------------
__global__ __launch_bounds__(256) void final_topk_kernel(const float* __restrict__ cls_scores,
                                                         const float* __restrict__ top_boxes,
                                                         float* __restrict__ out) {
    int b   = blockIdx.x;
    int tid = threadIdx.x;
    __shared__ float rv[256];
    __shared__ int   ri[256];
    __shared__ unsigned taken[FK / 32];   // 1280 words

    for (int i = tid; i < FK / 32; i += 256) taken[i] = 0u;
    __syncthreads();

    const float* cs = cls_scores + (size_t)b * FK;
    float* obox = out;                                   // B*100*4
    float* osc  = out + (size_t)B_ * MAXDET * 4;         // B*100
    int*   ocls = (int*)(out + (size_t)B_ * MAXDET * 5); // B*100
    int*   oval = (int*)(out + (size_t)B_ * MAXDET * 6); // B

    int validCount = 0;
    for (int iter = 0; iter < MAXDET; ++iter) {
        float bv = -INFINITY;
        int   bi = 0x7FFFFFFF;
        for (int i = tid; i < FK; i += 256) {
            if ((taken[i >> 5] >> (i & 31)) & 1u) continue;
            float v = cs[i];
            if (v > bv || (v == bv && i < bi)) { bv = v; bi = i; }
        }
        rv[tid] = bv; ri[tid] = bi;
        __syncthreads();
        for (int s = 128; s > 0; s >>= 1) {
            if (tid < s) {
                float ov2 = rv[tid + s]; int oi = ri[tid + s];
                if (ov2 > rv[tid] || (ov2 == rv[tid] && oi < ri[tid])) {
                    rv[tid] = ov2; ri[tid] = oi;
                }
            }
            __syncthreads();
        }
        float s0 = rv[0];
        int   fi = ri[0];
        if (tid == 0) {
            if (fi >= 0 && fi < FK) taken[fi >> 5] |= 1u << (fi & 31);
            bool valid = (s0 != -INFINITY);
            int cls = fi / KTOP;
            size_t bsrc = ((size_t)b * FK + fi) * 4;
            size_t odst = ((size_t)b * MAXDET + iter) * 4;
            obox[odst + 0] = valid ? top_boxes[bsrc + 0] : 0.f;
            obox[odst + 1] = valid ? top_boxes[bsrc + 1] : 0.f;
            obox[odst + 2] = valid ? top_boxes[bsrc + 2] : 0.f;
            obox[odst + 3] = valid ? top_boxes[bsrc + 3] : 0.f;
            osc[(size_t)b * MAXDET + iter]  = valid ? s0 : 0.f;
            ocls[(size_t)b * MAXDET + iter] = valid ? cls : 0;
            if (valid) validCount++;
        }
        __syncthreads();
    }
    if (tid == 0) oval[b] = validCount;
}

// ---------------------------------------------------------------------------
extern "C" void kernel_launch(void* const* d_in, const int* in_sizes, int n_in,
                              void* d_out, int out_size, void* d_ws, size_t ws_size,
                              hipStream_t stream) {
    (void)in_sizes; (void)n_in; (void)out_size;
    const float* boxes  = (const float*)d_in[0];  // (B, N, 1, 4)
    const float* scores = (const float*)d_in[1];  // (B, N, C)
    float* out = (float*)d_out;
    char*  ws  = (char*)d_ws;

    size_t off = 0;
    float* top_scores = (float*)(ws + off); off += (size_t)B_ * NCLS * KTOP * 4;
    float* top_boxes  = (float*)(ws + off); off += (size_t)B_ * NCLS * KTOP * 4 * 4;
    float* cls_scores = (float*)(ws + off); off += (size_t)B_ * NCLS * KTOP * 4;

    const float* sbase = scores;
    int strideN = NCLS;  // original (B,N,C) layout, stride C between anchors
    size_t needT = off + (size_t)B_ * NCLS * N_ANCH * 4;
    if (ws_size >= needT) {
        float* st = (float*)(ws + off);
        dim3 tg((N_ANCH + 31) / 32, (NCLS + 31) / 32, B_);
        transpose_kernel<<<tg, 256, 0, stream>>>(scores, st);
        sbase = st;
        strideN = 1;  // (B,C,N) layout, unit stride
    }

    topk_kernel<<<B_ * NCLS, 256, 0, stream>>>(sbase, strideN, boxes, top_scores, top_boxes);
    nms_kernel<<<B_ * NCLS, 256, 0, stream>>>(top_scores, top_boxes, cls_scores);
    final_topk_kernel<<<B_, 256, 0, stream>>>(cls_scores, top_boxes, out);
}